// Net_46729244180871
// MI455X (gfx1250) — compile-verified
//
#include <hip/hip_runtime.h>

typedef __attribute__((ext_vector_type(2))) float v2f;
typedef __attribute__((ext_vector_type(8))) float v8f;

#define NN 1000000
#define NE 4000000
#define NG 50000
#define IND 9
#define HID 16

// ---------------- init output to -inf ----------------
__global__ void k_init_out(float* __restrict__ out, int n) {
    int i = blockIdx.x * blockDim.x + threadIdx.x;
    if (i < n) out[i] = __int_as_float(0xFF800000); // -inf
}

// ---------------- layer-1 node transform via f32 WMMA ----------------
// One wave handles a tile of 16 nodes. A = x[tile] (16x9, K padded to 12),
// B = W (9x16 padded). Three V_WMMA_F32_16X16X4_F32 per weight matrix.
// Produces t1 = x@W1n and acc1 = x@W1r + b1 (accumulator seed for edge scatter).
__global__ void k_node_l1(const float* __restrict__ x,
                          const float* __restrict__ W1n,
                          const float* __restrict__ W1r,
                          const float* __restrict__ b1,
                          float* __restrict__ t1,
                          float* __restrict__ acc1) {
    int tile = (blockIdx.x * blockDim.x + threadIdx.x) >> 5; // one 16-node tile per wave
    if (tile * 16 >= NN) return;                              // wave-uniform exit, EXEC stays full
    int lane = threadIdx.x & 31;
    int half = lane >> 4;   // 0: lanes 0-15, 1: lanes 16-31
    int m    = lane & 15;   // A row within tile
    int col  = lane & 15;   // B/C column (hidden dim)

    v8f cn = {};
    v8f cr = {};
    float bb = b1[col];
#pragma unroll
    for (int v = 0; v < 8; ++v) cr[v] = bb;

#pragma unroll
    for (int j = 0; j < 3; ++j) {
        int k0 = 4 * j + 2 * half;          // this lane's two K indices: k0, k0+1
        int node = tile * 16 + m;
        v2f a, bn, br;
        a.x  = (k0     < IND && node < NN) ? x[node * IND + k0]     : 0.f;
        a.y  = (k0 + 1 < IND && node < NN) ? x[node * IND + k0 + 1] : 0.f;
        bn.x = (k0     < IND) ? W1n[k0 * HID + col]       : 0.f;
        bn.y = (k0 + 1 < IND) ? W1n[(k0 + 1) * HID + col] : 0.f;
        br.x = (k0     < IND) ? W1r[k0 * HID + col]       : 0.f;
        br.y = (k0 + 1 < IND) ? W1r[(k0 + 1) * HID + col] : 0.f;
        cn = __builtin_amdgcn_wmma_f32_16x16x4_f32(false, a, false, bn, (short)0, cn, false, false);
        cr = __builtin_amdgcn_wmma_f32_16x16x4_f32(false, a, false, br, (short)0, cr, false, false);
    }

    // C layout: VGPR v -> row (v + 8*half), column = lane&15
#pragma unroll
    for (int v = 0; v < 8; ++v) {
        int node = tile * 16 + v + 8 * half;
        if (node < NN) {
            t1[node * HID + col]   = cn[v];
            acc1[node * HID + col] = cr[v];
        }
    }
}

// ---------------- layer-1 edge scatter: 16 lanes per edge ----------------
__global__ void k_edge_l1(const int* __restrict__ ei,        // [src(NE), dst(NE)]
                          const float* __restrict__ ea,      // [NE,3]
                          const float* __restrict__ W1e,     // [3,16]
                          const float* __restrict__ t1,
                          float* __restrict__ acc1) {
    long gid = (long)blockIdx.x * blockDim.x + threadIdx.x;
    long e = gid >> 4;
    if (e >= NE) return;
    int h = (int)(gid & 15);
    int s = ei[e];
    int d = ei[NE + e];
    float msg = t1[(long)s * HID + h]
              + ea[e * 3 + 0] * W1e[0 * HID + h]
              + ea[e * 3 + 1] * W1e[1 * HID + h]
              + ea[e * 3 + 2] * W1e[2 * HID + h];
    atomicAdd(&acc1[(long)d * HID + h], msg);
}

// ---------------- relu + both layer-2 node projections (fused) ----------------
__global__ void k_node_mid(const float* __restrict__ acc1,
                           const float* __restrict__ W2n,   // [16]
                           const float* __restrict__ W2r,   // [16]
                           const float* __restrict__ b2,    // [1]
                           float* __restrict__ s2,          // [NN] relu(h)@W2n
                           float* __restrict__ h2) {        // [NN] seeded with relu(h)@W2r + b2
    int i = blockIdx.x * blockDim.x + threadIdx.x;
    if (i >= NN) return;
    const float4* p = (const float4*)(acc1 + (long)i * HID);
    float sn = 0.f, sr = 0.f;
#pragma unroll
    for (int q = 0; q < 4; ++q) {
        float4 v = p[q];
        float r0 = fmaxf(v.x, 0.f), r1 = fmaxf(v.y, 0.f);
        float r2 = fmaxf(v.z, 0.f), r3 = fmaxf(v.w, 0.f);
        sn += r0 * W2n[q * 4 + 0] + r1 * W2n[q * 4 + 1] + r2 * W2n[q * 4 + 2] + r3 * W2n[q * 4 + 3];
        sr += r0 * W2r[q * 4 + 0] + r1 * W2r[q * 4 + 1] + r2 * W2r[q * 4 + 2] + r3 * W2r[q * 4 + 3];
    }
    s2[i] = sn;
    h2[i] = sr + b2[0];
}

// ---------------- layer-2 edge scatter (scalar per edge) ----------------
__global__ void k_edge_l2(const int* __restrict__ ei,
                          const float* __restrict__ ea,
                          const float* __restrict__ W2e,    // [3]
                          const float* __restrict__ s2,
                          float* __restrict__ h2) {
    int e = blockIdx.x * blockDim.x + threadIdx.x;
    if (e >= NE) return;
    int s = ei[e];
    int d = ei[NE + e];
    float msg = s2[s] + ea[e * 3 + 0] * W2e[0] + ea[e * 3 + 1] * W2e[1] + ea[e * 3 + 2] * W2e[2];
    atomicAdd(&h2[d], msg);
}

// ---------------- global max pool per graph ----------------
__device__ __forceinline__ void atomicMaxF(float* addr, float v) {
    if (v >= 0.f) atomicMax((int*)addr, __float_as_int(v));
    else          atomicMin((unsigned int*)addr, __float_as_uint(v));
}

__global__ void k_pool(const float* __restrict__ h2,
                       const int* __restrict__ batch,
                       float* __restrict__ out) {
    int i = blockIdx.x * blockDim.x + threadIdx.x;
    if (i >= NN) return;
    atomicMaxF(&out[batch[i]], h2[i]);
}

extern "C" void kernel_launch(void* const* d_in, const int* in_sizes, int n_in,
                              void* d_out, int out_size, void* d_ws, size_t ws_size,
                              hipStream_t stream) {
    const float* x    = (const float*)d_in[0];
    const int*   ei   = (const int*)  d_in[1];   // [2, NE] flat: src then dst
    const float* ea   = (const float*)d_in[2];
    const int*   batch= (const int*)  d_in[3];
    const float* W1n  = (const float*)d_in[4];
    const float* W1e  = (const float*)d_in[5];
    const float* W1r  = (const float*)d_in[6];
    const float* b1   = (const float*)d_in[7];
    const float* W2n  = (const float*)d_in[8];
    const float* W2e  = (const float*)d_in[9];
    const float* W2r  = (const float*)d_in[10];
    const float* b2   = (const float*)d_in[11];
    float* out = (float*)d_out;

    float* ws   = (float*)d_ws;
    float* t1   = ws;                         // [NN*16]  x@W1n
    float* acc1 = ws + (size_t)NN * HID;      // [NN*16]  x@W1r+b1, then += edge msgs
    // after layer 1, t1's space is dead -> reuse it for layer-2 scalars
    float* s2   = t1;                         // [NN]
    float* h2   = t1 + NN;                    // [NN]

    // 1) out = -inf
    k_init_out<<<(out_size + 255) / 256, 256, 0, stream>>>(out, out_size);
    // 2) node transform (WMMA): 62500 tiles, 8 waves/block
    {
        int waves = (NN + 15) / 16;
        int threads = waves * 32;
        k_node_l1<<<(threads + 255) / 256, 256, 0, stream>>>(x, W1n, W1r, b1, t1, acc1);
    }
    // 3) layer-1 edge scatter: 16 threads per edge
    {
        long threads = (long)NE * 16;
        k_edge_l1<<<(int)((threads + 255) / 256), 256, 0, stream>>>(ei, ea, W1e, t1, acc1);
    }
    // 4) relu + layer-2 node projections
    k_node_mid<<<(NN + 255) / 256, 256, 0, stream>>>(acc1, W2n, W2r, b2, s2, h2);
    // 5) layer-2 edge scatter
    k_edge_l2<<<(NE + 255) / 256, 256, 0, stream>>>(ei, ea, W2e, s2, h2);
    // 6) per-graph max pool
    k_pool<<<(NN + 255) / 256, 256, 0, stream>>>(h2, batch, out);
}